// ScaledDotProductAttention_54563264528840
// MI455X (gfx1250) — compile-verified
//
#include <hip/hip_runtime.h>

typedef __attribute__((ext_vector_type(2))) float v2f;
typedef __attribute__((ext_vector_type(8))) float v8f;

#define S_LEN 1024
#define DK    64
#define NHEAD 8
#define NBATCH 8

// One wave (32 threads) handles a 16-query tile against all 1024 keys.
__global__ __launch_bounds__(32)
void attn_wmma_f32_kernel(const float* __restrict__ Q,
                          const float* __restrict__ K,
                          const float* __restrict__ V,
                          const float* __restrict__ Msk,
                          float* __restrict__ ctx_out,
                          float* __restrict__ sc_out)
{
    // 16 rows x 1024 cols of unnormalized exp-scores: exactly 64 KB LDS
    __shared__ float lds_s[16 * S_LEN];

    const int lane = threadIdx.x & 31;
    const int n    = lane & 15;   // column within 16-wide tile / row id for A-frags
    const int hi   = lane >> 4;   // which 16-lane half

    const int wg = blockIdx.x;
    const int qt = wg % (S_LEN / 16);
    const int bh = wg / (S_LEN / 16);
    const int b  = bh / NHEAD;
    const int q0 = qt * 16;

    const float* Qb = Q   + (size_t)bh * S_LEN * DK;
    const float* Kb = K   + (size_t)bh * S_LEN * DK;
    const float* Vb = V   + (size_t)bh * S_LEN * DK;
    const float* Mb = Msk + (size_t)b  * S_LEN;

    // ---- Load Q tile as 16 A-fragments (16x4 f32 each, 2 VGPRs per frag) ----
    // A layout: lanes 0-15 hold K=0,1 in (x,y); lanes 16-31 hold K=2,3.
    v2f qa[16];
    {
        const float* qrow = Qb + (size_t)(q0 + n) * DK + 2 * hi;
        #pragma unroll
        for (int kk = 0; kk < 16; ++kk)
            qa[kk] = *(const v2f*)(qrow + 4 * kk);
    }

    v8f rs  = {};                      // per-lane rowsum partials (rows r+8*hi)
    v8f ctx[4] = {{}, {}, {}, {}};     // context accumulators for d-tiles 0..3

    #pragma unroll 1
    for (int kt = 0; kt < S_LEN / 16; ++kt) {
        // ---- QK^T for this 16-key tile: 16 WMMAs over K-dim (D=64, 4 each) ----
        const float* krow = Kb + (size_t)(kt * 16 + n) * DK + 2 * hi;
        v8f c = {};
        #pragma unroll
        for (int kk = 0; kk < 16; ++kk) {
            v2f kb = *(const v2f*)(krow + 4 * kk);
            c = __builtin_amdgcn_wmma_f32_16x16x4_f32(
                    false, qa[kk], false, kb, (short)0, c, false, false);
        }

        // ---- s = exp(c * 1/sqrt(64)) * mask ; accumulate rowsums; spill to LDS ----
        const float mval = Mb[kt * 16 + n];
        #pragma unroll
        for (int r = 0; r < 8; ++r) {
            float s = __expf(c[r] * 0.125f) * mval;   // raw exp, as in reference
            rs[r] += s;
            lds_s[(r + 8 * hi) * S_LEN + kt * 16 + n] = s;
        }
        __syncthreads();  // single-wave WG: barrier is a NOP, orders LDS use

        // ---- Re-stripe S tile from C-layout to A-layout via LDS ----
        v2f sa[4];
        #pragma unroll
        for (int kk2 = 0; kk2 < 4; ++kk2)
            sa[kk2] = *(const v2f*)&lds_s[n * S_LEN + kt * 16 + 4 * kk2 + 2 * hi];

        // ---- ctx += S_tile (16x16) x V_tile (16x64): 4 d-tiles x 4 K-steps ----
        #pragma unroll
        for (int dt = 0; dt < 4; ++dt) {
            #pragma unroll
            for (int kk2 = 0; kk2 < 4; ++kk2) {
                const int key0 = kt * 16 + 4 * kk2 + 2 * hi;
                v2f vb;
                vb.x = Vb[(size_t)key0       * DK + dt * 16 + n];
                vb.y = Vb[(size_t)(key0 + 1) * DK + dt * 16 + n];
                ctx[dt] = __builtin_amdgcn_wmma_f32_16x16x4_f32(
                              false, sa[kk2], false, vb, (short)0, ctx[dt], false, false);
            }
        }
        __syncthreads();
    }

    // ---- Finish rowsums: reduce across the 16 lanes of each half (wave32) ----
    #pragma unroll
    for (int r = 0; r < 8; ++r) {
        float v = rs[r];
        v += __shfl_xor(v, 1, 32);
        v += __shfl_xor(v, 2, 32);
        v += __shfl_xor(v, 4, 32);
        v += __shfl_xor(v, 8, 32);
        rs[r] = v;  // full rowsum for row r + 8*hi, replicated in each half
    }

    // Broadcast all 16 row inverses to every lane (rows 0-7 from lane 0, 8-15 from lane 16)
    float invm[16];
    #pragma unroll
    for (int r = 0; r < 8; ++r) {
        invm[r]     = __shfl(rs[r], 0, 32);
        invm[r + 8] = __shfl(rs[r], 16, 32);
    }
    #pragma unroll
    for (int m = 0; m < 16; ++m)
        invm[m] = 1.0f / (invm[m] + 1e-8f);

    // ---- Store context: ctx[dt][r] is row (r+8*hi), col (dt*16 + n) ----
    {
        float* cout = ctx_out + ((size_t)bh * S_LEN + q0) * DK;
        #pragma unroll
        for (int r = 0; r < 8; ++r) {
            const int m = r + 8 * hi;
            const float inv = 1.0f / (rs[r] + 1e-8f);
            #pragma unroll
            for (int dt = 0; dt < 4; ++dt)
                cout[(size_t)m * DK + dt * 16 + n] = ctx[dt][r] * inv;
        }
    }

    // ---- Normalize + store scores: the one pass over the 256MB output ----
    // Coalesced: each lane owns 4 consecutive cols -> b128 LDS load + b128 global store
    {
        float* sout = sc_out + (size_t)bh * S_LEN * S_LEN + (size_t)q0 * S_LEN;
        #pragma unroll 1
        for (int m = 0; m < 16; ++m) {
            const float inv = invm[m];
            #pragma unroll
            for (int cb = 0; cb < S_LEN; cb += 128) {
                const int col = cb + lane * 4;
                const float* lp = &lds_s[m * S_LEN + col];
                float v0 = lp[0] * inv, v1 = lp[1] * inv, v2 = lp[2] * inv, v3 = lp[3] * inv;
                float* gp = &sout[(size_t)m * S_LEN + col];
                gp[0] = v0; gp[1] = v1; gp[2] = v2; gp[3] = v3;
            }
        }
    }
}

extern "C" void kernel_launch(void* const* d_in, const int* in_sizes, int n_in,
                              void* d_out, int out_size, void* d_ws, size_t ws_size,
                              hipStream_t stream) {
    const float* Q = (const float*)d_in[0];
    const float* K = (const float*)d_in[1];
    const float* V = (const float*)d_in[2];
    const float* M = (const float*)d_in[3];

    float* out     = (float*)d_out;
    float* ctx_out = out;                                          // (B,H,S,D)
    float* sc_out  = out + (size_t)NBATCH * NHEAD * S_LEN * DK;    // (B,H,S,S)

    const int nblocks = NBATCH * NHEAD * (S_LEN / 16);  // 4096 waves
    attn_wmma_f32_kernel<<<nblocks, 32, 0, stream>>>(Q, K, V, M, ctx_out, sc_out);
}